// GCNTwoOrder_48524540510799
// MI455X (gfx1250) — compile-verified
//
#include <hip/hip_runtime.h>
#include <math.h>

typedef __attribute__((ext_vector_type(2))) float v2f;
typedef __attribute__((ext_vector_type(8))) float v8f;

#define WMMA_F32X4(a, b, c) \
  __builtin_amdgcn_wmma_f32_16x16x4_f32(false, (a), false, (b), (short)0, (c), false, false)

// ---------------- small utility kernels ----------------

__global__ void zero_kernel(float* __restrict__ p, long long n) {
  long long i = (long long)blockIdx.x * blockDim.x + threadIdx.x;
  if (i < n) p[i] = 0.0f;
}

__global__ void init_ones2_kernel(float* __restrict__ a, float* __restrict__ b, int n) {
  int i = blockIdx.x * blockDim.x + threadIdx.x;
  if (i < n) { a[i] = 1.0f; b[i] = 1.0f; }   // self-loop contributes degree 1
}

__global__ void deg_kernel(const int* __restrict__ dst, float* __restrict__ deg, long long E) {
  long long i = (long long)blockIdx.x * blockDim.x + threadIdx.x;
  if (i < E) atomicAdd(&deg[dst[i]], 1.0f);
}

__global__ void rsqrt2_kernel(float* __restrict__ a, float* __restrict__ b, int n) {
  int i = blockIdx.x * blockDim.x + threadIdx.x;
  if (i < n) {
    a[i] = rsqrtf(fmaxf(a[i], 1.0f));
    b[i] = rsqrtf(fmaxf(b[i], 1.0f));
  }
}

// ---------------- GEMM 1: P = X @ [W11 | W21]  (K=512, N=32) ----------------
// One wave computes a 16x32 output tile via fp32 WMMA (16x16x4).
// A layout (16x4 f32): lane&15 = M, lane>>4 selects K-pair {0,1} vs {2,3}.
// B layout (4x16 f32): lane&15 = N, lane>>4 selects K-pair.
// C layout: VGPR r -> row r (lanes 0-15) / row r+8 (lanes 16-31), N = lane&15.

__global__ __launch_bounds__(256) void gemm_x_kernel(
    const float* __restrict__ X, const float* __restrict__ W11,
    const float* __restrict__ W21, float* __restrict__ P, int n) {
  const int wave = (blockIdx.x * blockDim.x + threadIdx.x) >> 5;
  const int lane = threadIdx.x & 31;
  const int lo = lane & 15, hi = lane >> 4;
  const int tiles = (n + 15) >> 4;
  if (wave >= tiles) return;                      // uniform per wave: EXEC stays all-1
  const int row  = wave * 16 + lo;
  const int arow = row < n ? row : n - 1;
  const float* xr = X + (long long)arow * 512;

  v8f c0 = {};   // cols 0..15  (W11)
  v8f c1 = {};   // cols 16..31 (W21)

  #pragma unroll 4
  for (int kk = 0; kk < 512; kk += 4) {
    const int k0 = kk + 2 * hi;
    v2f a;
    a.x = __builtin_nontemporal_load(xr + k0);     // X streamed once: NT, keep L2 for scatter tables
    a.y = __builtin_nontemporal_load(xr + k0 + 1);
    v2f b0, b1;
    b0.x = W11[k0 * 16 + lo];  b0.y = W11[(k0 + 1) * 16 + lo];
    b1.x = W21[k0 * 16 + lo];  b1.y = W21[(k0 + 1) * 16 + lo];
    c0 = WMMA_F32X4(a, b0, c0);
    c1 = WMMA_F32X4(a, b1, c1);
  }

  #pragma unroll
  for (int r = 0; r < 8; ++r) {
    const int prow = wave * 16 + r + 8 * hi;
    if (prow < n) {
      P[(long long)prow * 32 + lo]      = c0[r];
      P[(long long)prow * 32 + 16 + lo] = c1[r];
    }
  }
}

// ---------------- edge aggregation (always 16 channels now): ----------------
// out[d][ch] += in[s][ch] * dinv[s]*dinv[d]
// 16 consecutive lanes handle one edge's channels -> coalesced gather + same-line atomics.

__global__ __launch_bounds__(256) void agg16_kernel(
    const int* __restrict__ src, const int* __restrict__ dst,
    const float* __restrict__ dinv, const float* __restrict__ in, int ldin,
    float* __restrict__ out, long long E) {
  long long gid = (long long)blockIdx.x * blockDim.x + threadIdx.x;
  long long e = gid >> 4;
  int ch = (int)(gid & 15);
  if (e >= E) return;
  const int s = src[e], d = dst[e];
  const float w = dinv[s] * dinv[d];
  atomicAdd(&out[(long long)d * 16 + ch], in[(long long)s * ldin + ch] * w);
}

// ---------------- epilogue layer 1: H = elu(agg + P*dinv^2 + b), C=16 ----------------

__global__ __launch_bounds__(256) void finish_elu_kernel(
    const float* __restrict__ agg, const float* __restrict__ P, int ldp,
    const float* __restrict__ dinv, const float* __restrict__ b,
    float* __restrict__ H, int n) {
  long long t = (long long)blockIdx.x * blockDim.x + threadIdx.x;
  int i = (int)(t >> 4), j = (int)(t & 15);
  if (i < n) {
    const float di = dinv[i];
    const float x = agg[t] + P[(long long)i * ldp + j] * di * di + b[j];
    H[t] = x > 0.0f ? x : expm1f(x);
  }
}

// ---------------- GEMM 2 (fused layer-2 epilogue):
// C = (agg + H*dinv^2) @ W + b    (K=16, N=40 -> 3 masked 16-col tiles)
// Aggregation commutes with the right-multiply, so we aggregated H in 16-dim
// and project 16->40 here, folding the self-loop term into the A operand.

__global__ __launch_bounds__(256) void gemm_l2_kernel(
    const float* __restrict__ agg, const float* __restrict__ H,
    const float* __restrict__ dinv, const float* __restrict__ W,
    const float* __restrict__ b, float* __restrict__ Cout, int n) {
  const int wave = (blockIdx.x * blockDim.x + threadIdx.x) >> 5;
  const int lane = threadIdx.x & 31;
  const int lo = lane & 15, hi = lane >> 4;
  const int tiles = (n + 15) >> 4;
  if (wave >= tiles) return;
  const int row  = wave * 16 + lo;
  const int arow = row < n ? row : n - 1;
  const float di = dinv[arow];
  const float di2 = di * di;
  const float* ar = agg + (long long)arow * 16;
  const float* hr = H   + (long long)arow * 16;

  v8f acc[3] = {{}, {}, {}};
  #pragma unroll
  for (int kk = 0; kk < 16; kk += 4) {
    const int k0 = kk + 2 * hi;
    v2f av = *(const v2f*)(ar + k0);
    v2f hv = *(const v2f*)(hr + k0);
    v2f a;
    a.x = fmaf(hv.x, di2, av.x);                  // fused self-loop contribution
    a.y = fmaf(hv.y, di2, av.y);
    #pragma unroll
    for (int ct = 0; ct < 3; ++ct) {
      const int col = ct * 16 + lo;
      v2f bb;
      bb.x = col < 40 ? W[k0 * 40 + col]       : 0.0f;
      bb.y = col < 40 ? W[(k0 + 1) * 40 + col] : 0.0f;
      acc[ct] = WMMA_F32X4(a, bb, acc[ct]);
    }
  }
  #pragma unroll
  for (int r = 0; r < 8; ++r) {
    const int orow = wave * 16 + r + 8 * hi;
    if (orow < n) {
      #pragma unroll
      for (int ct = 0; ct < 3; ++ct) {
        const int col = ct * 16 + lo;
        if (col < 40) Cout[(long long)orow * 40 + col] = acc[ct][r] + b[col];
      }
    }
  }
}

// ---------------- GEMM 3: out = [C1|C2] @ Wc + bc  (K=80, N=40) ----------------

__global__ __launch_bounds__(256) void gemm_out_kernel(
    const float* __restrict__ C1, const float* __restrict__ C2,
    const float* __restrict__ Wc, const float* __restrict__ bc,
    float* __restrict__ out, int n) {
  const int wave = (blockIdx.x * blockDim.x + threadIdx.x) >> 5;
  const int lane = threadIdx.x & 31;
  const int lo = lane & 15, hi = lane >> 4;
  const int tiles = (n + 15) >> 4;
  if (wave >= tiles) return;
  const int row  = wave * 16 + lo;
  const int arow = row < n ? row : n - 1;

  v8f acc[3] = {{}, {}, {}};
  #pragma unroll
  for (int kk = 0; kk < 80; kk += 4) {            // K pairs never straddle the 40-boundary
    const int k0 = kk + 2 * hi;
    const float* A = (kk < 40) ? C1 : C2;
    const int kq = (kk < 40) ? k0 : k0 - 40;
    v2f a = *(const v2f*)(A + (long long)arow * 40 + kq);
    #pragma unroll
    for (int ct = 0; ct < 3; ++ct) {
      const int col = ct * 16 + lo;
      v2f b;
      b.x = col < 40 ? Wc[k0 * 40 + col]       : 0.0f;
      b.y = col < 40 ? Wc[(k0 + 1) * 40 + col] : 0.0f;
      acc[ct] = WMMA_F32X4(a, b, acc[ct]);
    }
  }
  #pragma unroll
  for (int r = 0; r < 8; ++r) {
    const int orow = wave * 16 + r + 8 * hi;
    if (orow < n) {
      #pragma unroll
      for (int ct = 0; ct < 3; ++ct) {
        const int col = ct * 16 + lo;
        if (col < 40) out[(long long)orow * 40 + col] = acc[ct][r] + bc[col];
      }
    }
  }
}

// ---------------- row-wise log_softmax over 40 cols, one wave32 per row ----------------

__global__ __launch_bounds__(256) void logsoftmax_kernel(float* __restrict__ out, int n) {
  const int wave = (blockIdx.x * blockDim.x + threadIdx.x) >> 5;
  const int lane = threadIdx.x & 31;
  if (wave >= n) return;
  float* row = out + (long long)wave * 40;
  const float v0 = row[lane];
  const float v1 = (lane < 8) ? row[32 + lane] : -INFINITY;
  float m = fmaxf(v0, v1);
  #pragma unroll
  for (int off = 16; off; off >>= 1) m = fmaxf(m, __shfl_xor(m, off, 32));
  float e = expf(v0 - m) + ((lane < 8) ? expf(v1 - m) : 0.0f);
  #pragma unroll
  for (int off = 16; off; off >>= 1) e += __shfl_xor(e, off, 32);
  const float lse = m + logf(e);
  row[lane] = v0 - lse;
  if (lane < 8) row[32 + lane] = v1 - lse;
}

// ---------------- launch ----------------

extern "C" void kernel_launch(void* const* d_in, const int* in_sizes, int n_in,
                              void* d_out, int out_size, void* d_ws, size_t ws_size,
                              hipStream_t stream) {
  const float* X    = (const float*)d_in[0];
  const int*   adj1 = (const int*)  d_in[1];
  const int*   adj2 = (const int*)  d_in[2];
  const float* W11  = (const float*)d_in[3];
  const float* b11  = (const float*)d_in[4];
  const float* W12  = (const float*)d_in[5];
  const float* b12  = (const float*)d_in[6];
  const float* W21  = (const float*)d_in[7];
  const float* b21  = (const float*)d_in[8];
  const float* W22  = (const float*)d_in[9];
  const float* b22  = (const float*)d_in[10];
  const float* Wc   = (const float*)d_in[11];
  const float* bc   = (const float*)d_in[12];
  float* out = (float*)d_out;

  const int       N  = in_sizes[0] / 512;
  const long long E1 = in_sizes[1] / 2;
  const long long E2 = in_sizes[2] / 2;
  const int* src1 = adj1;          const int* dst1 = adj1 + E1;
  const int* src2 = adj2;          const int* dst2 = adj2 + E2;

  // workspace carve-up (all fp32)
  float* w = (float*)d_ws;
  long long off = 0;
  float* dinv1 = w + off; off += N;
  float* dinv2 = w + off; off += N;
  float* P     = w + off; off += (long long)N * 32;   // [P1 | P2], ld = 32
  float* H1    = w + off; off += (long long)N * 16;
  float* H2    = w + off; off += (long long)N * 16;
  float* agg16 = w + off; off += (long long)N * 16;   // reused serially 4x
  float* C1b   = w + off; off += (long long)N * 40;
  float* C2b   = w + off; off += (long long)N * 40;

  const int TB = 256;
  const int nb_N   = (N + TB - 1) / TB;
  const int gemm_b = (((N + 15) / 16) * 32 + TB - 1) / TB;   // one wave per 16-row tile
  const long long n16 = (long long)N * 16;
  const int nb_16  = (int)((n16 + TB - 1) / TB);
  const int eb1    = (int)((E1 * 16 + TB - 1) / TB);
  const int eb2    = (int)((E2 * 16 + TB - 1) / TB);

  // degrees -> dinv (self-loop baked in as 1.0)
  init_ones2_kernel<<<nb_N, TB, 0, stream>>>(dinv1, dinv2, N);
  deg_kernel<<<(int)((E1 + TB - 1) / TB), TB, 0, stream>>>(dst1, dinv1, E1);
  deg_kernel<<<(int)((E2 + TB - 1) / TB), TB, 0, stream>>>(dst2, dinv2, E2);
  rsqrt2_kernel<<<nb_N, TB, 0, stream>>>(dinv1, dinv2, N);

  // P = X @ [W11 | W21]   (fp32 WMMA, memory-bound on streaming X)
  gemm_x_kernel<<<gemm_b, TB, 0, stream>>>(X, W11, W21, P, N);

  // branch 1, layer 1: aggregate P1 (16ch), epilogue -> H1
  zero_kernel<<<nb_16, TB, 0, stream>>>(agg16, n16);
  agg16_kernel<<<eb1, TB, 0, stream>>>(src1, dst1, dinv1, P, 32, agg16, E1);
  finish_elu_kernel<<<nb_16, TB, 0, stream>>>(agg16, P, 32, dinv1, b11, H1, N);

  // branch 2, layer 1: aggregate P2 (16ch), epilogue -> H2
  zero_kernel<<<nb_16, TB, 0, stream>>>(agg16, n16);
  agg16_kernel<<<eb2, TB, 0, stream>>>(src2, dst2, dinv2, P + 16, 32, agg16, E2);
  finish_elu_kernel<<<nb_16, TB, 0, stream>>>(agg16, P + 16, 32, dinv2, b21, H2, N);

  // branch 1, layer 2: aggregate H1 in 16-dim, then fused (agg + self-loop) @ W12 + b12
  zero_kernel<<<nb_16, TB, 0, stream>>>(agg16, n16);
  agg16_kernel<<<eb1, TB, 0, stream>>>(src1, dst1, dinv1, H1, 16, agg16, E1);
  gemm_l2_kernel<<<gemm_b, TB, 0, stream>>>(agg16, H1, dinv1, W12, b12, C1b, N);

  // branch 2, layer 2
  zero_kernel<<<nb_16, TB, 0, stream>>>(agg16, n16);
  agg16_kernel<<<eb2, TB, 0, stream>>>(src2, dst2, dinv2, H2, 16, agg16, E2);
  gemm_l2_kernel<<<gemm_b, TB, 0, stream>>>(agg16, H2, dinv2, W22, b22, C2b, N);

  // combine + log_softmax
  gemm_out_kernel<<<gemm_b, TB, 0, stream>>>(C1b, C2b, Wc, bc, out, N);
  logsoftmax_kernel<<<(int)(((long long)N * 32 + TB - 1) / TB), TB, 0, stream>>>(out, N);
}